// VectorQuantizer_85693187489816
// MI455X (gfx1250) — compile-verified
//
#include <hip/hip_runtime.h>

typedef __attribute__((ext_vector_type(2))) float v2f;
typedef __attribute__((ext_vector_type(8))) float v8f;

#define VQ_K   1024
#define VQ_D   64
#define VQ_HW  1024            // H*W
#define KC     256             // codes staged in LDS per chunk
#define EPAD   68              // padded LDS row stride (floats): 16B-aligned, bank-friendly
#define Q_SIZE (64*64*32*32)   // B*D*H*W
#define LOSS_OFF (Q_SIZE)
#define IDX_OFF  (Q_SIZE + 1)

__global__ __launch_bounds__(256) void vq_main_kernel(
    const float* __restrict__ z, const float* __restrict__ emb,
    float* __restrict__ out, float* __restrict__ wsPartial)
{
  __shared__ float sE[KC * EPAD];
  __shared__ float sNorm[KC];
  __shared__ int   sIdx[8 * 16];
  __shared__ float sLoss[8];

  const int tid  = threadIdx.x;
  const int lane = tid & 31;
  const int wave = tid >> 5;
  const int lo   = lane & 15;
  const int hi   = lane >> 4;

  // wave handles rows [rowBase, rowBase+16)
  const int rowBase = blockIdx.x * 128 + wave * 16;
  const int n  = rowBase + lo;
  const int bb = n >> 10;          // batch index (H*W = 1024)
  const int hw = n & 1023;
  const float* zb = z + (size_t)bb * (VQ_D * VQ_HW) + hw;

  // A fragments: ISA 16x4 f32 A layout. step s covers d = 4s..4s+3:
  //   a.x = z[row=lo][4s + 2*hi], a.y = z[row=lo][4s + 2*hi + 1]
  v2f afrag[16];
#pragma unroll
  for (int s = 0; s < 16; ++s) {
    const int d = 4 * s + 2 * hi;
    afrag[s].x = zb[d * VQ_HW];
    afrag[s].y = zb[(d + 1) * VQ_HW];
  }

  float best[8];
  int   bidx[8];
#pragma unroll
  for (int v = 0; v < 8; ++v) { best[v] = 3.4e38f; bidx[v] = 0; }

  for (int c = 0; c < VQ_K / KC; ++c) {
    __syncthreads();
    // cooperative stage: 256 codes x 64 f32 into padded LDS rows (float4, coalesced)
    const float4* src = (const float4*)(emb + c * KC * VQ_D);
    for (int i = tid; i < KC * VQ_D / 4; i += 256) {
      float4 q = src[i];
      const int code = i >> 4;            // 16 float4 per code
      const int dd   = (i & 15) << 2;
      float* dst = &sE[code * EPAD + dd];
      dst[0] = q.x; dst[1] = q.y; dst[2] = q.z; dst[3] = q.w;
    }
    __syncthreads();
    { // ||e||^2 per staged code (one code per thread)
      float a = 0.f;
      const float* row = &sE[tid * EPAD];
#pragma unroll 8
      for (int d = 0; d < VQ_D; ++d) a += row[d] * row[d];
      sNorm[tid] = a;
    }
    __syncthreads();

    for (int t = 0; t < KC / 16; ++t) {
      v8f acc0 = {};
      v8f acc1 = {};
      // B fragments: 4x16 f32 layout mirrors A: b.x = e[col=lo][4s+2*hi], b.y = +1
      const float* ebase = &sE[(t * 16 + lo) * EPAD + 2 * hi];
#pragma unroll
      for (int s = 0; s < 16; s += 2) {
        v2f b0 = *(const v2f*)(ebase + 4 * s);
        v2f b1 = *(const v2f*)(ebase + 4 * (s + 1));
        acc0 = __builtin_amdgcn_wmma_f32_16x16x4_f32(
            false, afrag[s],     false, b0, (short)0, acc0, false, false);
        acc1 = __builtin_amdgcn_wmma_f32_16x16x4_f32(
            false, afrag[s + 1], false, b1, (short)0, acc1, false, false);
      }
      const float en   = sNorm[t * 16 + lo];
      const int   code = c * KC + t * 16 + lo;
#pragma unroll
      for (int v = 0; v < 8; ++v) {
        // dist ~ ||e||^2 - 2 z.e (||z||^2 constant per row -> dropped for argmin)
        const float sv = en - 2.0f * (acc0[v] + acc1[v]);
        if (sv < best[v]) { best[v] = sv; bidx[v] = code; }
      }
    }
  }

  // argmin across the 16 lanes of each half (C layout: lane lo holds col lo; row = v + 8*hi)
#pragma unroll
  for (int m = 1; m <= 8; m <<= 1) {
#pragma unroll
    for (int v = 0; v < 8; ++v) {
      const float ob = __shfl_xor(best[v], m, 32);
      const int   oi = __shfl_xor(bidx[v], m, 32);
      if (ob < best[v] || (ob == best[v] && oi < bidx[v])) { best[v] = ob; bidx[v] = oi; }
    }
  }

  if (lo == 0) { // lanes 0 (rows 0..7) and 16 (rows 8..15)
#pragma unroll
    for (int v = 0; v < 8; ++v) sIdx[wave * 16 + hi * 8 + v] = bidx[v];
  }
  __syncthreads();

  const int myIdx = sIdx[wave * 16 + lo];
  if (hi == 0) out[IDX_OFF + n] = (float)myIdx;   // indices emitted as float values

  // gather codebook row, write quantized (straight-through == quantized numerically),
  // accumulate sum((q - z)^2) using the register-held z fragments
  const float* erow = emb + (size_t)myIdx * VQ_D;
  float* qb = out + (size_t)bb * (VQ_D * VQ_HW) + hw;
  float lpart = 0.f;
#pragma unroll
  for (int s = 0; s < 16; ++s) {
    const int d = 4 * s + 2 * hi;
    const float q0 = erow[d];
    const float q1 = erow[d + 1];
    qb[d * VQ_HW]       = q0;
    qb[(d + 1) * VQ_HW] = q1;
    const float d0 = q0 - afrag[s].x;
    const float d1 = q1 - afrag[s].y;
    lpart += d0 * d0 + d1 * d1;
  }
#pragma unroll
  for (int m = 1; m <= 16; m <<= 1) lpart += __shfl_xor(lpart, m, 32);
  if (lane == 0) sLoss[wave] = lpart;
  __syncthreads();
  if (tid == 0) {
    float tot = 0.f;
#pragma unroll
    for (int w2 = 0; w2 < 8; ++w2) tot += sLoss[w2];
    wsPartial[blockIdx.x] = tot;
  }
}

// deterministic fixed-order final reduction: loss = (1 + beta) * sum / B
__global__ __launch_bounds__(256) void vq_loss_kernel(
    const float* __restrict__ part, float* __restrict__ out)
{
  __shared__ float s[256];
  const int t = threadIdx.x;
  s[t] = part[t] + part[t + 256];   // 512 block partials
  __syncthreads();
  for (int ofs = 128; ofs > 0; ofs >>= 1) {
    if (t < ofs) s[t] += s[t + ofs];
    __syncthreads();
  }
  if (t == 0) out[LOSS_OFF] = 1.25f * s[0] / 64.0f;
}

extern "C" void kernel_launch(void* const* d_in, const int* in_sizes, int n_in,
                              void* d_out, int out_size, void* d_ws, size_t ws_size,
                              hipStream_t stream) {
  const float* z   = (const float*)d_in[0];   // (64, 64, 32, 32) f32
  const float* emb = (const float*)d_in[1];   // (1024, 64) f32
  float* out = (float*)d_out;                 // [quantized | loss | indices]
  float* ws  = (float*)d_ws;                  // 512 block partial losses

  vq_main_kernel<<<512, 256, 0, stream>>>(z, emb, out, ws);
  vq_loss_kernel<<<1, 256, 0, stream>>>(ws, out);
}